// NeuralODE_60181081751904
// MI455X (gfx1250) — compile-verified
//
#include <hip/hip_runtime.h>
#include <hip/hip_bf16.h>

// Neural ODE Euler rollout on CDNA5 (gfx1250), f32 WMMA 16x16x4.
//
// Shapes: B=4096, T=512, D=16, H=64.
// One wave per 16 batch rows (256 single-wave blocks -> one wave per WGP).
// State x lives in WMMA C/D-layout registers across all 511 steps;
// W1/W2/b1/b2 preloaded once into WMMA B-layout registers.
// Layer1: A=[x | I_k | 0pad] (16x20) x W1ext (20x64) -> 4 Ntiles x 5 Ktiles WMMA
// Layer2: A=tanh(h) (16x64)  x W2   (64x16)          -> 16 Ktiles WMMA
// LDS only for the per-step D-layout -> A-layout transposes (padded strides
// 20 / 68 dwords: conflict-free on 64 banks, b64-aligned loads).
// tanh via hardware V_TANH_F32 (or a branch-free exp2/rcp fallback) so the
// whole loop body is EXEC-uniform (WMMA requires EXEC all-ones).

typedef float v2f __attribute__((ext_vector_type(2)));
typedef float v8f __attribute__((ext_vector_type(8)));

#define TT 512
#define DD 16
#define HH 64
#define XSTRIDE 20     // x tile row stride (dwords), padded for bank spread
#define HSTRIDE 68     // h tile row stride (dwords), padded for bank spread

static __device__ __forceinline__ float fast_tanh(float x) {
#if __has_builtin(__builtin_amdgcn_tanhf)
    return __builtin_amdgcn_tanhf(x);           // v_tanh_f32
#else
    // branch-free: tanh(x) = copysign(1 - 2e/(1+e), x), e = exp(-2|x|)
    const float ax = __builtin_fabsf(x);
  #if __has_builtin(__builtin_amdgcn_exp2f)
    const float e = __builtin_amdgcn_exp2f(ax * -2.8853900817779268f); // -2/ln2... (=-2*log2(e))
  #else
    const float e = __expf(-2.0f * ax);
  #endif
  #if __has_builtin(__builtin_amdgcn_rcpf)
    const float r = __builtin_amdgcn_rcpf(1.0f + e);
  #else
    const float r = 1.0f / (1.0f + e);
  #endif
    return __builtin_copysignf(__builtin_fmaf(-2.0f * e, r, 1.0f), x);
#endif
}

__global__ __launch_bounds__(32) void node_euler_wmma(
    const float* __restrict__ x0,   // [B, D]
    const float* __restrict__ cur,  // [B, T]
    const float* __restrict__ t,    // [T]
    const float* __restrict__ W1,   // [D+1, H] row-major
    const float* __restrict__ b1,   // [H]
    const float* __restrict__ W2,   // [H, D] row-major
    const float* __restrict__ b2,   // [D]
    float* __restrict__ out)        // [B, T, D]
{
    __shared__ float s_t[TT];
    __shared__ float s_x[16 * XSTRIDE];
    __shared__ float s_h[16 * HSTRIDE];

    const int lane = threadIdx.x & 31;
    const int lo   = lane & 15;       // N index / row index within half
    const int hi   = lane >> 4;       // 0 = lanes 0-15, 1 = lanes 16-31
    const int tileBase = blockIdx.x * 16;

    // stage time grid (single wave: uniform trip count, barrier ops are NOPs)
    for (int i = lane; i < TT; i += 32) s_t[i] = t[i];
    __syncthreads();

    // ---- preload weights into WMMA B-layout registers (reused 511x) ----
    // B layout (4x16 f32, 2 VGPRs): lane holds col N = lo;
    //   v.x -> K = 4*kt + 2*hi, v.y -> K = 4*kt + 2*hi + 1
    v2f  w1b[4][5];
    float b1v[4];
    for (int nt = 0; nt < 4; ++nt) {
        const int n = nt * 16 + lo;
        b1v[nt] = b1[n];
        for (int kt = 0; kt < 5; ++kt) {
            const int k0 = 4 * kt + 2 * hi;
            v2f v;
            v.x = (k0     < DD + 1) ? W1[k0 * HH + n]       : 0.0f;
            v.y = (k0 + 1 < DD + 1) ? W1[(k0 + 1) * HH + n] : 0.0f;
            w1b[nt][kt] = v;
        }
    }
    v2f w2b[16];
    for (int kt = 0; kt < 16; ++kt) {
        const int k0 = 4 * kt + 2 * hi;
        v2f v;
        v.x = W2[k0 * DD + lo];
        v.y = W2[(k0 + 1) * DD + lo];
        w2b[kt] = v;
    }
    const float b2v = b2[lo];

    // ---- state x in WMMA C/D layout: xd[r] = x[M = r + 8*hi][N = lo] ----
    float xd[8];
    #pragma unroll
    for (int r = 0; r < 8; ++r)
        xd[r] = x0[(size_t)(tileBase + r + 8 * hi) * DD + lo];

    // out[:, 0, :] = x0
    #pragma unroll
    for (int r = 0; r < 8; ++r)
        out[(size_t)(tileBase + r + 8 * hi) * (TT * DD) + lo] = xd[r];

    const float* curRow = cur + (size_t)(tileBase + lo) * TT;
    float tk = s_t[0];

    #pragma unroll 1
    for (int k = 0; k < TT - 1; ++k) {
        // per-tile current sample I_k (row = lo), dt from time grid
        const float ik  = curRow[k];
        const float tk1 = s_t[k + 1];
        const float dt  = tk1 - tk;
        tk = tk1;

        // ---- transpose x: D-layout regs -> LDS -> A-layout regs ----
        #pragma unroll
        for (int r = 0; r < 8; ++r)
            s_x[(r + 8 * hi) * XSTRIDE + lo] = xd[r];
        __builtin_amdgcn_wave_barrier();          // same-wave LDS is in-order
        v2f xa[5];
        #pragma unroll
        for (int kt = 0; kt < 4; ++kt)            // aligned b64: cols {K,K+1}
            xa[kt] = *(const v2f*)&s_x[lo * XSTRIDE + 4 * kt + 2 * hi];
        { v2f v; v.x = hi ? 0.0f : ik; v.y = 0.0f; xa[4] = v; }  // K=16 row
        __builtin_amdgcn_wave_barrier();

        // ---- layer 1: h = tanh([x|I|0] @ W1 + b1), store D-layout to LDS ----
        #pragma unroll
        for (int nt = 0; nt < 4; ++nt) {
            v8f a;
            #pragma unroll
            for (int e = 0; e < 8; ++e) a[e] = b1v[nt];
            #pragma unroll
            for (int kt = 0; kt < 5; ++kt)
                a = __builtin_amdgcn_wmma_f32_16x16x4_f32(
                        false, xa[kt], false, w1b[nt][kt],
                        (short)0, a, false, false);
            #pragma unroll
            for (int r = 0; r < 8; ++r)
                s_h[(r + 8 * hi) * HSTRIDE + nt * 16 + lo] = fast_tanh(a[r]);
        }
        __builtin_amdgcn_wave_barrier();

        // ---- layer 2: dx = h @ W2 + b2 (A tiles streamed from LDS) ----
        v8f acc;
        #pragma unroll
        for (int e = 0; e < 8; ++e) acc[e] = b2v;
        #pragma unroll
        for (int kt = 0; kt < 16; ++kt) {
            v2f ha = *(const v2f*)&s_h[lo * HSTRIDE + 4 * kt + 2 * hi];
            acc = __builtin_amdgcn_wmma_f32_16x16x4_f32(
                      false, ha, false, w2b[kt],
                      (short)0, acc, false, false);
        }
        __builtin_amdgcn_wave_barrier();

        // ---- Euler update + trajectory store (64B per half-wave per row) ----
        #pragma unroll
        for (int r = 0; r < 8; ++r) {
            xd[r] = __builtin_fmaf(dt, acc[r], xd[r]);
            out[(size_t)(tileBase + r + 8 * hi) * (TT * DD)
                + (size_t)(k + 1) * DD + lo] = xd[r];
        }
    }
}

extern "C" void kernel_launch(void* const* d_in, const int* in_sizes, int n_in,
                              void* d_out, int out_size, void* d_ws, size_t ws_size,
                              hipStream_t stream) {
    const float* x0  = (const float*)d_in[0];   // [B, D]
    const float* cur = (const float*)d_in[1];   // [B, T]
    const float* t   = (const float*)d_in[2];   // [T]
    const float* W1  = (const float*)d_in[3];   // [D+1, H]
    const float* b1  = (const float*)d_in[4];   // [H]
    const float* W2  = (const float*)d_in[5];   // [H, D]
    const float* b2  = (const float*)d_in[6];   // [D]
    float* out = (float*)d_out;                 // [B, T, D]

    const int B = in_sizes[0] / DD;             // 4096
    const int blocks = B / 16;                  // 256 single-wave blocks
    node_euler_wmma<<<blocks, 32, 0, stream>>>(x0, cur, t, W1, b1, W2, b2, out);
}